// SpikingPCLayer_32770600468654
// MI455X (gfx1250) — compile-verified
//
#include <hip/hip_runtime.h>
#include <hip/hip_bf16.h>
#include <math.h>

// ---------------- problem constants (match reference) ----------------
#define IN_DIM   128
#define HID_DIM  128
#define HEADS    4
#define D_HEAD   32
#define T_STEPS  25
#define BETA     0.9f
#define THRESH   1.0f
#define NEG_SLOPE 0.2f

typedef float v2f __attribute__((ext_vector_type(2)));
typedef float v8f __attribute__((ext_vector_type(8)));

// ---------------------------------------------------------------------
// Kernel 0: init accumulators. d_out <- 0, e_max <- -inf, denom <- 0
// ---------------------------------------------------------------------
__global__ void init_kernel(float* __restrict__ out, float* __restrict__ emax,
                            float* __restrict__ denom, int n_out, int n_nh) {
    int i = blockIdx.x * blockDim.x + threadIdx.x;
    if (i < n_out) out[i] = 0.0f;
    if (i < n_nh) { emax[i] = -INFINITY; denom[i] = 0.0f; }
}

// ---------------------------------------------------------------------
// Kernel 1: xw = x @ W  via V_WMMA_F32_16X16X4_F32 (fp32 matrix core path)
// Block = 256 threads = 8 waves. Block b covers rows [16b,16b+16);
// wave w covers cols [16w,16w+16). K loop: 32 steps of 4.
// A layout (ISA 7.12.2, 32-bit A 16x4): lanes 0-15 hold M=lane,
//   VGPR0=K0,VGPR1=K1; lanes 16-31 hold K2/K3.  B symmetric with N=lane.
// C/D: VGPR v -> M = v + 8*(lane>>4), N = lane&15.
// ---------------------------------------------------------------------
__global__ __launch_bounds__(256) void gemm_xw_kernel(
    const float* __restrict__ X, const float* __restrict__ W,
    float* __restrict__ XW, int nrows) {
    const int wave = threadIdx.x >> 5;
    const int lane = threadIdx.x & 31;
    const int half = lane >> 4;      // 0 or 1
    const int l15  = lane & 15;
    const int r0   = blockIdx.x * 16;
    const int c0   = wave * 16;
    if (r0 >= nrows) return;

    const float* __restrict__ xrow = X + (size_t)(r0 + l15) * IN_DIM;
    v8f acc = {};
    #pragma unroll 4
    for (int kk = 0; kk < IN_DIM / 4; ++kk) {
        const int kb = kk * 4 + half * 2;      // this half-wave's K pair
        v2f a, b;
        a.x = xrow[kb];
        a.y = xrow[kb + 1];
        b.x = W[(size_t)kb * HID_DIM + c0 + l15];
        b.y = W[(size_t)(kb + 1) * HID_DIM + c0 + l15];
        acc = __builtin_amdgcn_wmma_f32_16x16x4_f32(
                /*neg_a=*/false, a, /*neg_b=*/false, b,
                /*c_mod=*/(short)0, acc, /*reuse_a=*/false, /*reuse_b=*/false);
    }
    float* __restrict__ op = XW + (size_t)(r0 + half * 8) * HID_DIM + c0 + l15;
    #pragma unroll
    for (int v = 0; v < 8; ++v) op[(size_t)v * HID_DIM] = acc[v];
}

// ---------------------------------------------------------------------
// Kernel 2: attention logits per (node, head):
//   a_src[n,h] = dot(xw[n,h,:], att_src[h,:]);  same for a_dst.
// ---------------------------------------------------------------------
__global__ void attn_logits_kernel(const float* __restrict__ XW,
                                   const float* __restrict__ att_src,
                                   const float* __restrict__ att_dst,
                                   float* __restrict__ a_src,
                                   float* __restrict__ a_dst, int n_nodes) {
    int i = blockIdx.x * blockDim.x + threadIdx.x;   // i = n*HEADS + h
    if (i >= n_nodes * HEADS) return;
    int h = i & (HEADS - 1);
    const float* __restrict__ xwp = XW + (size_t)(i >> 2) * HID_DIM + h * D_HEAD;
    const float* __restrict__ as  = att_src + h * D_HEAD;
    const float* __restrict__ ad  = att_dst + h * D_HEAD;
    float ss = 0.0f, sd = 0.0f;
    #pragma unroll
    for (int d = 0; d < D_HEAD; ++d) {
        float v = xwp[d];
        ss = fmaf(v, as[d], ss);
        sd = fmaf(v, ad[d], sd);
    }
    a_src[i] = ss;
    a_dst[i] = sd;
}

// float atomic max via int/uint ordering trick (init value is -inf)
__device__ __forceinline__ void atomicMaxF32(float* addr, float val) {
    if (val >= 0.0f) atomicMax((int*)addr, __float_as_int(val));
    else             atomicMin((unsigned int*)addr, (unsigned int)__float_as_int(val));
}

// ---------------------------------------------------------------------
// Kernel 3: edge pass A — e = leaky_relu(a_src[s]+a_dst[d]); segment max.
// Edges [0,E) from edge_index; [E, E+N) are self loops (s=d=idx-E).
// ---------------------------------------------------------------------
__global__ void edge_max_kernel(const int* __restrict__ ei, int E, int n_nodes,
                                const float* __restrict__ a_src,
                                const float* __restrict__ a_dst,
                                float* __restrict__ ebuf,
                                float* __restrict__ emax) {
    int idx = blockIdx.x * blockDim.x + threadIdx.x;
    int total = E + n_nodes;
    if (idx >= total) return;
    int s, d;
    if (idx < E) { s = ei[idx]; d = ei[E + idx]; }
    else         { s = idx - E; d = s; }
    #pragma unroll
    for (int h = 0; h < HEADS; ++h) {
        float v = a_src[s * HEADS + h] + a_dst[d * HEADS + h];
        v = (v > 0.0f) ? v : NEG_SLOPE * v;
        ebuf[(size_t)idx * HEADS + h] = v;
        atomicMaxF32(&emax[d * HEADS + h], v);
    }
}

// ---------------------------------------------------------------------
// Kernel 4: edge pass B — e_exp = exp(e - emax[dst]); segment sum.
// (overwrites ebuf with e_exp)
// ---------------------------------------------------------------------
__global__ void edge_exp_kernel(const int* __restrict__ ei, int E, int n_nodes,
                                float* __restrict__ ebuf,
                                const float* __restrict__ emax,
                                float* __restrict__ denom) {
    int idx = blockIdx.x * blockDim.x + threadIdx.x;
    int total = E + n_nodes;
    if (idx >= total) return;
    int d = (idx < E) ? ei[E + idx] : (idx - E);
    #pragma unroll
    for (int h = 0; h < HEADS; ++h) {
        float v = __expf(ebuf[(size_t)idx * HEADS + h] - emax[d * HEADS + h]);
        ebuf[(size_t)idx * HEADS + h] = v;
        atomicAdd(&denom[d * HEADS + h], v);   // global_atomic_add_f32 (L2-resident)
    }
}

// ---------------------------------------------------------------------
// Kernel 5: edge pass C — out[dst] += xw[src] * alpha. One wave per edge;
// lane l moves features [4l, 4l+4) (head = l>>3) as a float4 + 4 fp32 atomics.
// xw (51 MB) and out (51 MB) are L2-resident on MI455X (192 MB L2).
// ---------------------------------------------------------------------
__global__ __launch_bounds__(256) void edge_aggregate_kernel(
    const int* __restrict__ ei, int E, int n_nodes,
    const float* __restrict__ XW, const float* __restrict__ ebuf,
    const float* __restrict__ denom, float* __restrict__ out) {
    int edge = blockIdx.x * (blockDim.x >> 5) + (threadIdx.x >> 5);
    int lane = threadIdx.x & 31;
    int total = E + n_nodes;
    if (edge >= total) return;
    int s, d;
    if (edge < E) { s = ei[edge]; d = ei[E + edge]; }
    else          { s = edge - E; d = s; }
    int h = lane >> 3;                                   // (lane*4)>>5
    float alpha = ebuf[(size_t)edge * HEADS + h] /
                  (denom[d * HEADS + h] + 1e-16f);
    const float4 xv = ((const float4*)(XW + (size_t)s * HID_DIM))[lane];
    float* op = out + (size_t)d * HID_DIM + lane * 4;
    atomicAdd(op + 0, xv.x * alpha);
    atomicAdd(op + 1, xv.y * alpha);
    atomicAdd(op + 2, xv.z * alpha);
    atomicAdd(op + 3, xv.w * alpha);
}

// ---------------------------------------------------------------------
// Kernel 6: LIF accumulation over T steps + residual, in place on d_out.
//   h = gat_out + bias; mem=0; T times: mem=beta*mem+h; spike&reset.
//   out = (sum spk)/T + x
// ---------------------------------------------------------------------
__global__ void lif_residual_kernel(float* __restrict__ out,
                                    const float* __restrict__ x,
                                    const float* __restrict__ bias, int n) {
    int i = blockIdx.x * blockDim.x + threadIdx.x;
    if (i >= n) return;
    float h = out[i] + bias[i & (HID_DIM - 1)];
    float mem = 0.0f, cnt = 0.0f;
    #pragma unroll
    for (int t = 0; t < T_STEPS; ++t) {
        mem = fmaf(BETA, mem, h);
        float spk = (mem > THRESH) ? 1.0f : 0.0f;
        mem -= spk * THRESH;
        cnt += spk;
    }
    out[i] = cnt * (1.0f / (float)T_STEPS) + x[i];
}

// ---------------------------------------------------------------------
extern "C" void kernel_launch(void* const* d_in, const int* in_sizes, int n_in,
                              void* d_out, int out_size, void* d_ws, size_t ws_size,
                              hipStream_t stream) {
    const float* x       = (const float*)d_in[0];
    const int*   ei      = (const int*)d_in[1];     // [2, E]: row0=src, row1=dst
    const float* W       = (const float*)d_in[2];
    const float* att_src = (const float*)d_in[3];
    const float* att_dst = (const float*)d_in[4];
    const float* bias    = (const float*)d_in[5];
    float* out = (float*)d_out;

    const int n_nodes = in_sizes[0] / IN_DIM;       // 100000
    const int E       = in_sizes[1] / 2;            // 1600000
    const int total_e = E + n_nodes;                // + self loops

    // ---- workspace layout (floats), ~85 MB total ----
    float* xw    = (float*)d_ws;                    // n_nodes*HID_DIM
    float* a_src = xw    + (size_t)n_nodes * HID_DIM;   // n_nodes*HEADS
    float* a_dst = a_src + (size_t)n_nodes * HEADS;
    float* emax  = a_dst + (size_t)n_nodes * HEADS;
    float* denom = emax  + (size_t)n_nodes * HEADS;
    float* ebuf  = denom + (size_t)n_nodes * HEADS;     // total_e*HEADS
    (void)ws_size; (void)n_in; (void)out_size;

    const int n_out = n_nodes * HID_DIM;
    const int n_nh  = n_nodes * HEADS;

    // 0) init accumulators
    init_kernel<<<(n_out + 255) / 256, 256, 0, stream>>>(out, emax, denom, n_out, n_nh);

    // 1) xw = x @ W   (v_wmma_f32_16x16x4_f32; 8 waves/block = full 128 cols)
    gemm_xw_kernel<<<(n_nodes + 15) / 16, 256, 0, stream>>>(x, W, xw, n_nodes);

    // 2) attention logits
    attn_logits_kernel<<<(n_nh + 127) / 128, 128, 0, stream>>>(
        xw, att_src, att_dst, a_src, a_dst, n_nodes);

    // 3) per-target max (stable softmax)
    edge_max_kernel<<<(total_e + 255) / 256, 256, 0, stream>>>(
        ei, E, n_nodes, a_src, a_dst, ebuf, emax);

    // 4) exp + denominator
    edge_exp_kernel<<<(total_e + 255) / 256, 256, 0, stream>>>(
        ei, E, n_nodes, ebuf, emax, denom);

    // 5) weighted message aggregation (wave per edge)
    edge_aggregate_kernel<<<(total_e + 7) / 8, 256, 0, stream>>>(
        ei, E, n_nodes, xw, ebuf, denom, out);

    // 6) LIF over T steps + residual
    lif_residual_kernel<<<(n_out + 255) / 256, 256, 0, stream>>>(out, x, bias, n_out);
}